// GATedge_47218870452948
// MI455X (gfx1250) — compile-verified
//
#include <hip/hip_runtime.h>
#include <math.h>

typedef __attribute__((ext_vector_type(2))) float v2f;
typedef __attribute__((ext_vector_type(8))) float v8f;

#define BB 16      // batch
#define OO 1000    // operations
#define MM 100     // machines
#define MP 112     // MM padded to multiple of 16
#define FF 64      // feature dim
#define INS 8      // input feature dim
#define SLOPE 0.2f

__device__ __forceinline__ float leaky(float x) { return x > 0.f ? x : SLOPE * x; }

// out[r][f] = sum_i in[r*INS+i] * W[i*FF+f]   (tiny K=8 linear)
__global__ void k_linear(const float* __restrict__ in, const float* __restrict__ W,
                         float* __restrict__ out, int nrows) {
    int idx = blockIdx.x * blockDim.x + threadIdx.x;
    if (idx >= nrows * FF) return;
    int f = idx & (FF - 1);
    int r = idx >> 6;
    float acc = 0.f;
#pragma unroll
    for (int i = 0; i < INS; ++i) acc = fmaf(in[r * INS + i], W[i * FF + f], acc);
    out[idx] = acc;
}

// out[r] = sum_f feat[r*FF+f] * attn[f]
__global__ void k_rowdot(const float* __restrict__ feat, const float* __restrict__ attn,
                         float* __restrict__ out, int nrows) {
    int r = blockIdx.x * blockDim.x + threadIdx.x;
    if (r >= nrows) return;
    float acc = 0.f;
#pragma unroll 8
    for (int f = 0; f < FF; ++f) acc = fmaf(feat[r * FF + f], attn[f], acc);
    out[r] = acc;
}

// One block per (b, m): compute column max, then unnormalized exp scores into
// S_T[b][m][o] (M-major, rows >= MM zero-padded), plus denom / sum(S*edge).
__global__ void k_scores(const int* __restrict__ adj, const int* __restrict__ bidx,
                         const float* __restrict__ edge,
                         const float* __restrict__ el, const float* __restrict__ er,
                         const float* __restrict__ W_edge, const float* __restrict__ attn_l,
                         float* __restrict__ S, float* __restrict__ vmax,
                         float* __restrict__ den, float* __restrict__ esum) {
    int b = blockIdx.x;
    int m = blockIdx.y;
    int t = threadIdx.x;
    float* Srow = S + ((size_t)b * MP + m) * OO;
    if (m >= MM) {  // zero-pad rows so WMMA A-tiles never need predication
        for (int o = t; o < OO; o += blockDim.x) Srow[o] = 0.f;
        return;
    }
    __shared__ float red[256];
    __shared__ float s_c0;
    if (t == 0) {
        float c = 0.f;
        for (int f = 0; f < FF; ++f) c = fmaf(W_edge[f], attn_l[f], c);
        s_c0 = c;
    }
    __syncthreads();
    float c0 = s_c0;
    int ab = bidx[b];
    const int*   adjp  = adj  + (size_t)ab * OO * MM + m;
    const float* edgep = edge + (size_t)b  * OO * MM + m;
    const float* elp   = el + b * OO;
    float erm = er[b * MM + m];
    // pass 1: column max over valid entries
    float lmax = -1e30f;
    for (int o = t; o < OO; o += blockDim.x) {
        if (adjp[(size_t)o * MM] == 1) {
            float s = leaky(elp[o] + erm + edgep[(size_t)o * MM] * c0);
            lmax = fmaxf(lmax, s);
        }
    }
    red[t] = lmax; __syncthreads();
    for (int s = 128; s > 0; s >>= 1) { if (t < s) red[t] = fmaxf(red[t], red[t + s]); __syncthreads(); }
    float ekk = leaky(2.f * erm);
    float mx  = fmaxf(red[0], ekk);
    __syncthreads();
    // pass 2: unnormalized exp, denom, sum(S*edge)
    float ld = 0.f, le = 0.f;
    for (int o = t; o < OO; o += blockDim.x) {
        float ev = edgep[(size_t)o * MM];
        float sv = 0.f;
        if (adjp[(size_t)o * MM] == 1)
            sv = __expf(leaky(elp[o] + erm + ev * c0) - mx);
        Srow[o] = sv;
        ld += sv;
        le += sv * ev;
    }
    red[t] = ld; __syncthreads();
    for (int s = 128; s > 0; s >>= 1) { if (t < s) red[t] += red[t + s]; __syncthreads(); }
    ld = red[0]; __syncthreads();
    red[t] = le; __syncthreads();
    for (int s = 128; s > 0; s >>= 1) { if (t < s) red[t] += red[t + s]; __syncthreads(); }
    le = red[0];
    if (t == 0) {
        int i = b * MM + m;
        vmax[i] = mx;
        den[i]  = ld + __expf(ekk - mx);
        esum[i] = le;
    }
}

// grid (BB, MP/16), 128 threads = 4 waves; wave w owns f-tile [16w,16w+16).
// D(16m x 16f) += A(16m x 4o) * B(4o x 16f) via v_wmma_f32_16x16x4_f32, 250 K-steps.
__global__ void k_wmma(const float* __restrict__ S, const float* __restrict__ fsrc,
                       const float* __restrict__ fdst,
                       const float* __restrict__ er, const float* __restrict__ vmax,
                       const float* __restrict__ den, const float* __restrict__ esum,
                       const float* __restrict__ W_edge, float* __restrict__ out) {
    int b    = blockIdx.x;
    int mt   = blockIdx.y;
    int wave = threadIdx.x >> 5;
    int lane = threadIdx.x & 31;
    int kh   = lane >> 4;    // 0: K={0,1}, 1: K={2,3}
    int l16  = lane & 15;
    int f0   = wave * 16;

    // A: row m = mt*16 + l16, cols o0 + 2*kh + {0,1}  (contiguous -> v2f load)
    const float* Abase = S + ((size_t)b * MP + mt * 16 + l16) * OO + 2 * kh;
    // B: rows o0 + 2*kh + {0,1}, col f0 + l16
    const float* Bbase = fsrc + ((size_t)b * OO + 2 * kh) * FF + f0 + l16;

    v8f acc = {};
    for (int o0 = 0; o0 < OO; o0 += 4) {
        v2f a = *(const v2f*)(Abase + o0);
        v2f bv;
        bv.x = Bbase[(size_t)o0 * FF];
        bv.y = Bbase[(size_t)o0 * FF + FF];
        acc = __builtin_amdgcn_wmma_f32_16x16x4_f32(
            false, a, false, bv, (short)0, acc, false, false);
    }
    // epilogue: C/D layout -> VGPR j holds rows M = j + 8*kh, N = l16
#pragma unroll
    for (int j = 0; j < 8; ++j) {
        int m = mt * 16 + j + 8 * kh;
        if (m < MM) {
            int i = b * MM + m;
            float inv = 1.f / den[i];
            float akk = __expf(leaky(2.f * er[i]) - vmax[i]) * inv;
            int f = f0 + l16;
            float val = acc[j] * inv + esum[i] * inv * W_edge[f]
                      + fdst[(size_t)i * FF + f] * akk;
            out[(size_t)i * FF + f] = 1.f / (1.f + __expf(-val));
        }
    }
}

extern "C" void kernel_launch(void* const* d_in, const int* in_sizes, int n_in,
                              void* d_out, int out_size, void* d_ws, size_t ws_size,
                              hipStream_t stream) {
    const int*   adj   = (const int*)d_in[0];
    const int*   bidx  = (const int*)d_in[1];
    const float* ope   = (const float*)d_in[2];
    const float* ma    = (const float*)d_in[3];
    const float* edge  = (const float*)d_in[4];
    const float* Wsrc  = (const float*)d_in[5];
    const float* Wdst  = (const float*)d_in[6];
    const float* Wedge = (const float*)d_in[7];
    const float* al    = (const float*)d_in[8];
    const float* ar    = (const float*)d_in[9];
    float* out = (float*)d_out;

    float* ws   = (float*)d_ws;
    float* fs   = ws;                              // BB*OO*FF
    float* fd   = fs + (size_t)BB * OO * FF;       // BB*MM*FF
    float* el   = fd + (size_t)BB * MM * FF;       // BB*OO
    float* er   = el + (size_t)BB * OO;            // BB*MM
    float* S    = er + (size_t)BB * MM;            // BB*MP*OO
    float* vmax = S  + (size_t)BB * MP * OO;       // BB*MM
    float* den  = vmax + (size_t)BB * MM;          // BB*MM
    float* esum = den  + (size_t)BB * MM;          // BB*MM

    k_linear<<<(BB * OO * FF + 255) / 256, 256, 0, stream>>>(ope, Wsrc, fs, BB * OO);
    k_linear<<<(BB * MM * FF + 255) / 256, 256, 0, stream>>>(ma, Wdst, fd, BB * MM);
    k_rowdot<<<(BB * OO + 255) / 256, 256, 0, stream>>>(fs, al, el, BB * OO);
    k_rowdot<<<(BB * MM + 255) / 256, 256, 0, stream>>>(fd, ar, er, BB * MM);

    dim3 g2(BB, MP);
    k_scores<<<g2, 256, 0, stream>>>(adj, bidx, edge, el, er, Wedge, al, S, vmax, den, esum);

    dim3 g3(BB, MP / 16);
    k_wmma<<<g3, 128, 0, stream>>>(S, fs, fd, er, vmax, den, esum, Wedge, out);
}